// GHMC_4818953306441
// MI455X (gfx1250) — compile-verified
//
#include <hip/hip_runtime.h>
#include <hip/hip_bf16.h>

#define GHMC_BINS 10
#define BLOCK 256
#define WAVES_PER_BLOCK (BLOCK / 32)

typedef float v2f __attribute__((ext_vector_type(2)));
typedef float v8f __attribute__((ext_vector_type(8)));

// ---------------------------------------------------------------------------
// Pass 1: fused elementwise + per-bin accumulation.
// Per-wave private LDS histograms, conflict-free banking:
//   dword addr = waveBase + array*320 + bin*32 + lane
//   bank       = lane + 32*(bin&1)  -> all 32 lanes distinct banks, always.
// Emits per-workgroup partials row of 32 floats:
//   [0..9] = counts, [10..19] = bce sums, [20..31] = 0 (pad for WMMA reduce).
// ---------------------------------------------------------------------------
__global__ __launch_bounds__(BLOCK)
void ghmc_main_kernel(const float* __restrict__ pred,
                      const int* __restrict__ target,
                      const float* __restrict__ lweight,
                      float* __restrict__ partials,
                      long long nvec, long long total) {
    __shared__ float hist[WAVES_PER_BLOCK][2][GHMC_BINS * 32];

    const int tid  = threadIdx.x;
    const int wave = tid >> 5;
    const int lane = tid & 31;

    float* cnt = &hist[wave][0][0];
    float* sm  = &hist[wave][1][0];

    for (int i = tid; i < WAVES_PER_BLOCK * 2 * GHMC_BINS * 32; i += BLOCK)
        ((float*)hist)[i] = 0.0f;
    __syncthreads();

    const long long gstride = (long long)gridDim.x * BLOCK;
    const long long gtid    = (long long)blockIdx.x * BLOCK + tid;

    // Vectorized main loop: float4 / int4 (B128 global loads).
    for (long long i = gtid; i < nvec; i += gstride) {
        float4 p4 = ((const float4*)pred)[i];
        int4   t4 = ((const int4*)target)[i];
        float4 w4 = ((const float4*)lweight)[i];
#pragma unroll
        for (int j = 0; j < 4; ++j) {
            float p  = (&p4.x)[j];
            float tj = (float)((&t4.x)[j]);
            float w  = (&w4.x)[j];

            float validf = (w > 0.0f) ? 1.0f : 0.0f;
            float ap  = fabsf(p);
            float e   = __expf(-ap);                       // exp(-|p|), v_exp_f32
            float inv = __builtin_amdgcn_rcpf(1.0f + e);   // 1/(1+e)
            float s   = (p >= 0.0f) ? inv : e * inv;       // stable sigmoid
            float g   = fabsf(s - tj);                     // in [0,1]
            int   bin = (int)(g * 10.0f);
            bin = bin > (GHMC_BINS - 1) ? (GHMC_BINS - 1) : bin;

            float l1p = __logf(1.0f + e);                  // log1p(exp(-|p|))
            float bce = fmaxf(p, 0.0f) - p * tj + l1p;
            float wb  = validf * bce;

            int off = bin * 32 + lane;                     // conflict-free
            atomicAdd(&cnt[off], validf);                  // ds_add_f32
            atomicAdd(&sm[off],  wb);                      // ds_add_f32
        }
    }
    // Scalar tail (total not divisible by 4; not hit for this shape).
    for (long long i = nvec * 4 + gtid; i < total; i += gstride) {
        float p  = pred[i];
        float tj = (float)target[i];
        float w  = lweight[i];
        float validf = (w > 0.0f) ? 1.0f : 0.0f;
        float ap  = fabsf(p);
        float e   = __expf(-ap);
        float inv = __builtin_amdgcn_rcpf(1.0f + e);
        float s   = (p >= 0.0f) ? inv : e * inv;
        float g   = fabsf(s - tj);
        int   bin = (int)(g * 10.0f);
        bin = bin > (GHMC_BINS - 1) ? (GHMC_BINS - 1) : bin;
        float bce = fmaxf(p, 0.0f) - p * tj + __logf(1.0f + e);
        int off = bin * 32 + lane;
        atomicAdd(&cnt[off], validf);
        atomicAdd(&sm[off],  validf * bce);
    }
    __syncthreads();

    // Workgroup reduction: wave 0 folds waves then lanes, writes partial row.
    if (wave == 0) {
        for (int b = 0; b < GHMC_BINS; ++b) {
            float c = 0.0f, s2 = 0.0f;
#pragma unroll
            for (int w8 = 0; w8 < WAVES_PER_BLOCK; ++w8) {
                c  += hist[w8][0][b * 32 + lane];
                s2 += hist[w8][1][b * 32 + lane];
            }
#pragma unroll
            for (int d = 16; d > 0; d >>= 1) {
                c  += __shfl_xor(c,  d, 32);
                s2 += __shfl_xor(s2, d, 32);
            }
            if (lane == 0) {
                partials[(long long)blockIdx.x * 32 + b]      = c;
                partials[(long long)blockIdx.x * 32 + 10 + b] = s2;
            }
        }
        if (lane < 12) partials[(long long)blockIdx.x * 32 + 20 + lane] = 0.0f;
    }
}

// ---------------------------------------------------------------------------
// Pass 2: deterministic WMMA column-sum reduction of [P][32] partials.
// A = ones(16x4) f32 -> D[m,n] = sum_k B[k,n]; invariant to K striping.
// Wave w handles columns w*16 .. w*16+15. P is a multiple of 4.
// ---------------------------------------------------------------------------
__global__ __launch_bounds__(64)
void ghmc_finalize_kernel(const float* __restrict__ partials, int P,
                          float* __restrict__ out) {
    __shared__ float totals[32];
    const int tid  = threadIdx.x;
    const int wave = tid >> 5;
    const int lane = tid & 31;
    const int half = lane >> 4;
    const int col  = (lane & 15) + wave * 16;

    v2f a; a[0] = 1.0f; a[1] = 1.0f;   // ones A-matrix (any layout: all 1s)
    v8f c = {};
#pragma unroll 2
    for (int r = 0; r < P; r += 4) {
        // Each column n receives rows r..r+3 exactly once across halves/VGPRs;
        // column sums are invariant to which slot carries which K.
        v2f b;
        b[0] = partials[(long long)(r + 2 * half)     * 32 + col];
        b[1] = partials[(long long)(r + 2 * half + 1) * 32 + col];
        c = __builtin_amdgcn_wmma_f32_16x16x4_f32(
                /*neg_a=*/false, a, /*neg_b=*/false, b,
                /*c_mod=*/(short)0, c, /*reuse_a=*/false, /*reuse_b=*/false);
    }
    // D row 0, column = lane (documented C/D layout: N = lane for lanes 0-15).
    if (lane < 16) totals[col] = c[0];
    __syncthreads();

    if (tid == 0) {
        float n = 0.0f;
        for (int b = 0; b < GHMC_BINS; ++b)
            n += (totals[b] > 0.0f) ? 1.0f : 0.0f;
        n = fmaxf(n, 1.0f);
        float loss = 0.0f;
        for (int b = 0; b < GHMC_BINS; ++b)
            loss += totals[10 + b] / fmaxf(totals[b], 1.0f);
        out[0] = loss / n;   // LOSS_WEIGHT = 1.0
    }
}

extern "C" void kernel_launch(void* const* d_in, const int* in_sizes, int n_in,
                              void* d_out, int out_size, void* d_ws, size_t ws_size,
                              hipStream_t stream) {
    const float* pred    = (const float*)d_in[0];
    const int*   target  = (const int*)d_in[1];
    const float* lweight = (const float*)d_in[2];
    float*       out     = (float*)d_out;
    float*       parts   = (float*)d_ws;

    const long long total = (long long)in_sizes[0];   // N*C elements
    const long long nvec  = total / 4;

    // Workgroup count: enough to saturate HBM, bounded by workspace
    // (each WG needs one 32-float partial row), multiple of 4 for WMMA K.
    int maxWG = (int)(ws_size / (32 * sizeof(float)));
    int numWG = 4096;
    if (numWG > maxWG) numWG = maxWG;
    numWG &= ~3;
    if (numWG < 4) numWG = 4;

    ghmc_main_kernel<<<numWG, BLOCK, 0, stream>>>(pred, target, lweight,
                                                  parts, nvec, total);
    ghmc_finalize_kernel<<<1, 64, 0, stream>>>(parts, numWG, out);
}